// RecurrentLRSCell2_54365696032792
// MI455X (gfx1250) — compile-verified
//
#include <hip/hip_runtime.h>
#include <cmath>

typedef float v2f __attribute__((ext_vector_type(2)));
typedef float v8f __attribute__((ext_vector_type(8)));

#define BB 256   // batch
#define TT 512   // time steps
#define FF 256   // input features
#define DD 512   // state width
#define UU 128   // units per level

#define NBLK 32        // persistent blocks (must all be co-resident)
#define LDS_STRIDE 72  // 64 cols + 8 pad: lanes 0-15 / 16-31 hit disjoint banks

// ---------------------------------------------------------------------------
// Phase 1: xproj[r, :] = (x[r, :] - x[r-1, :]) @ K      (r = b*T + t)
// One wave computes a 16-row x 128-col strip: 1 A-fragment (16x4 f32) reused
// across 8 B-fragments per K-step.  Result written into d_out; the persistent
// phase-2 kernel reads it and overwrites in place with h_t.
//
// f32 WMMA fragment layout (ISA 7.12.2):
//   A 16x4:  lanes 0-15 -> rows M=lane, VGPR0=K0 VGPR1=K1; lanes 16-31 -> K2,K3
//   B 4x16:  lanes 0-15 -> cols N=lane, VGPR0=row K0, VGPR1=row K1;
//            lanes 16-31 -> rows K2,K3
//   C 16x16: VGPR v, lanes 0-15 -> (M=v, N=lane); lanes 16-31 -> (M=v+8, ...)
// ---------------------------------------------------------------------------
__global__ __launch_bounds__(128) void xproj_kernel(const float* __restrict__ x,
                                                    const float* __restrict__ W,
                                                    float* __restrict__ out) {
  const int lane   = threadIdx.x & 31;
  const int wave   = threadIdx.x >> 5;
  const int waveId = blockIdx.x * 4 + wave;      // 32768 waves total
  const int colGrp = waveId & 3;                 // 4 groups of 128 columns
  const int rowBlk = waveId >> 2;                // 8192 blocks of 16 rows
  const int m      = lane & 15;
  const int hlf    = lane >> 4;

  const int r = rowBlk * 16 + m;                 // global row = b*T + t
  const int t = r & (TT - 1);                    // T == 512 (power of 2)

  // Branchless dx: clamp prev-row pointer to a safe row, scale by 0/1.
  const float scale = (t != 0) ? 1.0f : 0.0f;
  const float* xrow  = x + (size_t)r * FF;
  const float* xprev = x + (size_t)((t != 0) ? r - 1 : r) * FF;

  v8f acc[8] = {};

  for (int k = 0; k < FF; k += 4) {
    const int ka = k + hlf * 2;
    v2f a;
    a.x = fmaf(-scale, xprev[ka],     xrow[ka]);
    a.y = fmaf(-scale, xprev[ka + 1], xrow[ka + 1]);

    const float* wp = W + (size_t)ka * DD + colGrp * 128 + m;
#pragma unroll
    for (int s = 0; s < 8; ++s) {
      v2f bfrag; bfrag.x = wp[s * 16]; bfrag.y = wp[s * 16 + DD];
      acc[s] = __builtin_amdgcn_wmma_f32_16x16x4_f32(
          false, a, false, bfrag, (short)0, acc[s], false, false);
    }
  }

#pragma unroll
  for (int s = 0; s < 8; ++s) {
#pragma unroll
    for (int v = 0; v < 8; ++v) {
      const int row = rowBlk * 16 + v + 8 * hlf;
      const int col = colGrp * 128 + s * 16 + m;
      out[(size_t)row * DD + col] = acc[s][v];
    }
  }
}

// ---------------------------------------------------------------------------
// Phase 2: ONE persistent kernel runs all 512 steps.
//   - 32 blocks x 128 threads (4 waves). block = (rowQuad, colGrp):
//       colGrp = blockIdx & 7  -> fixed 64 columns of D
//       rowQuad = blockIdx >> 3, wave -> 16 batch rows each
//   - Each block stages its 512x64 slice of R into LDS ONCE via CDNA5
//     async-to-LDS loads (global_load_async_to_lds_b32, ASYNCcnt), then all
//     B fragments come from ds_load (R never touches L2 again).
//   - Per step: rupd = (h1 - h2) @ R_slice (WMMA f32), fuse gate/state/tanh,
//     write h into d_out (overwriting xproj) and rotate state buffers.
//   - Device-wide barrier between steps: monotonic atomic counter,
//     release (__threadfence) before arrive, acquire load in spin, s_sleep.
// ---------------------------------------------------------------------------
__global__ __launch_bounds__(128) void recurrent_persistent(
    float* __restrict__ hA, float* __restrict__ hB, float* __restrict__ hC,
    float* __restrict__ s0, float* __restrict__ s1,
    const float* __restrict__ R, float* __restrict__ out,
    unsigned* __restrict__ counter) {
  extern __shared__ float ldsR[];                // [512][LDS_STRIDE]

  const int lane   = threadIdx.x & 31;
  const int wave   = threadIdx.x >> 5;
  const int colGrp = blockIdx.x & 7;             // 8 groups of 64 columns
  const int rowBlk = (blockIdx.x >> 3) * 4 + wave; // 16 blocks of 16 batch rows
  const int m      = lane & 15;
  const int hlf    = lane >> 4;
  const int d0     = colGrp * 64;
  const int b0     = rowBlk * 16;

  // ---- Stage this block's R[:, d0:d0+64] slice into LDS via async DMA ----
  for (int i = threadIdx.x; i < DD * 64; i += 128) {
    const int k = i >> 6;
    const int c = i & 63;
    const unsigned ldsByte = (unsigned)(size_t)(const void*)(ldsR + (k * LDS_STRIDE + c));
    const unsigned gByte   = (unsigned)(((unsigned)k * DD + (unsigned)(d0 + c)) * 4u);
    asm volatile("global_load_async_to_lds_b32 %0, %1, %2"
                 :: "v"(ldsByte), "v"(gByte), "s"(R) : "memory");
  }
  asm volatile("s_wait_asynccnt 0" ::: "memory");
  __syncthreads();

  float* hq[3] = {hA, hB, hC};
  float* sq[2] = {s0, s1};
  int i_new = 0, i_p1 = 1, i_p2 = 2;             // hB, hC zeroed -> h_{-1}=h_{-2}=0
  int i_sp = 0, i_sn = 1;                        // s0 zeroed -> s_{-1}=0

  const int ba = b0 + m;                         // A-fragment batch row

  for (int t = 0; t < TT; ++t) {
    const float* h1r = hq[i_p1] + (size_t)ba * DD;
    const float* h2r = hq[i_p2] + (size_t)ba * DD;
    const float* sprev = sq[i_sp];
    float* snew = sq[i_sn];
    float* hnew = hq[i_new];

    v8f acc[4] = {};
    for (int k = 0; k < DD; k += 4) {
      const int ka = k + hlf * 2;
      v2f a;
      a.x = h1r[ka]     - h2r[ka];
      a.y = h1r[ka + 1] - h2r[ka + 1];

      const float* lp = ldsR + ka * LDS_STRIDE + m;
#pragma unroll
      for (int s = 0; s < 4; ++s) {
        v2f bfrag;
        bfrag.x = lp[s * 16];
        bfrag.y = lp[s * 16 + LDS_STRIDE];
        acc[s] = __builtin_amdgcn_wmma_f32_16x16x4_f32(
            false, a, false, bfrag, (short)0, acc[s], false, false);
      }
    }

#pragma unroll
    for (int s = 0; s < 4; ++s) {
#pragma unroll
      for (int v = 0; v < 8; ++v) {
        const int b = b0 + v + 8 * hlf;          // batch index
        const int d = d0 + s * 16 + m;           // state column
        const size_t oidx = ((size_t)b * TT + t) * DD + d;
        const size_t sidx = (size_t)b * DD + d;

        const float upd  = acc[s][v] + out[oidx];                // + xproj
        const float gate = (d < UU) ? 1.0f : sprev[sidx - UU];
        const float st   = sprev[sidx] + gate * upd;
        snew[sidx] = st;
        const float h = tanhf(st);
        hnew[sidx] = h;
        out[oidx]  = h;
      }
    }

    // rotate h: new -> prev -> prev2 ; swap s
    { const int tmp = i_p2; i_p2 = i_p1; i_p1 = i_new; i_new = tmp; }
    { const int tmp = i_sp; i_sp = i_sn; i_sn = tmp; }

    // ---- device-wide step barrier (skip after final step) ----
    if (t != TT - 1) {
      __threadfence();                           // release this step's writes
      __syncthreads();
      if (threadIdx.x == 0) {
        atomicAdd(counter, 1u);
        const unsigned target = (unsigned)NBLK * (unsigned)(t + 1);
        while (__hip_atomic_load(counter, __ATOMIC_ACQUIRE,
                                 __HIP_MEMORY_SCOPE_AGENT) < target) {
          __builtin_amdgcn_s_sleep(2);
        }
      }
      __syncthreads();                           // WGP-shared acquire covers block
    }
  }
}

extern "C" void kernel_launch(void* const* d_in, const int* in_sizes, int n_in,
                              void* d_out, int out_size, void* d_ws, size_t ws_size,
                              hipStream_t stream) {
  (void)in_sizes; (void)n_in; (void)out_size; (void)ws_size;

  const float* x = (const float*)d_in[0];   // [B, T, F]
  const float* W = (const float*)d_in[1];   // [F, D]
  const float* R = (const float*)d_in[2];   // [D, D]
  float* out = (float*)d_out;               // [B, T, D]
  float* ws  = (float*)d_ws;

  const size_t BD = (size_t)BB * DD;
  float* hA = ws + 0 * BD;                  // rotating h buffers
  float* hB = ws + 1 * BD;                  // h_{t-1} at t=0 (zero)
  float* hC = ws + 2 * BD;                  // h_{t-2} at t=0 (zero)
  float* s0 = ws + 3 * BD;                  // s_{t-1} at t=0 (zero)
  float* s1 = ws + 4 * BD;
  unsigned* counter = (unsigned*)(ws + 5 * BD);

  // Zero state + barrier counter every call (deterministic across replays).
  hipMemsetAsync(ws, 0, 5 * BD * sizeof(float) + 64, stream);

  // Phase 1: parallel input projection into d_out.
  xproj_kernel<<<8192, 128, 0, stream>>>(x, W, out);

  // Phase 2: single persistent kernel, all 512 steps, grid-wide step barrier.
  // Dynamic LDS: 512 x LDS_STRIDE floats = 147456 B (fits 320 KB/WGP).
  recurrent_persistent<<<NBLK, 128, DD * LDS_STRIDE * sizeof(float), stream>>>(
      hA, hB, hC, s0, s1, R, out, counter);
}